// ScaledDotProductAttention_48713519071239
// MI455X (gfx1250) — compile-verified
//
#include <hip/hip_runtime.h>

// Problem constants (from the reference)
#define DE    1024      // d_embed
#define DH    64        // d_k = d_v
#define BATCH 4
#define SEQ   4096
#define NROW  (BATCH * SEQ)

typedef __attribute__((ext_vector_type(2))) float v2f;
typedef __attribute__((ext_vector_type(8))) float v8f;

// D = A(16x4 f32) * B(4x16 f32) + C(16x16 f32), one matrix per wave32.
__device__ __forceinline__ v8f wmma_f32(v2f a, v2f b, v8f c) {
  return __builtin_amdgcn_wmma_f32_16x16x4_f32(
      /*neg_a=*/false, a, /*neg_b=*/false, b,
      /*c_mod=*/(short)0, c, /*reuse_a=*/false, /*reuse_b=*/false);
}

// ---------------------------------------------------------------------------
// Kernel 1: Q/K/V projections.  [NROW x 1024] @ [1024 x 64]^T  (W is [64,1024])
// Block = 128 threads (4 waves). Wave w computes rows [rbase+16w, rbase+16w+16)
// of all three outputs. K-chunks of 32 staged in LDS (stride 36 = conflict-free
// for both the coalesced fill and the WMMA fragment reads).
// ---------------------------------------------------------------------------
#define KC 32
#define SX 36   // padded LDS row stride (floats)

__global__ __launch_bounds__(128) void qkv_proj_kernel(
    const float* __restrict__ x,
    const float* __restrict__ Wq, const float* __restrict__ Wk,
    const float* __restrict__ Wv,
    float* __restrict__ Qo, float* __restrict__ Ko, float* __restrict__ Vo)
{
  __shared__ float xs[64 * SX];
  __shared__ float wqs[64 * SX];
  __shared__ float wks[64 * SX];
  __shared__ float wvs[64 * SX];

  const int tid  = threadIdx.x;
  const int lane = tid & 31;
  const int w    = tid >> 5;
  const int rbase = blockIdx.x * 64;
  const int n15  = lane & 15;
  const int hi2  = (lane >> 4) << 1;   // 0 or 2  (A/B fragment K offset)
  const int hi8  = (lane >> 4) << 3;   // 0 or 8  (C/D fragment M offset)

  const v8f zero = {0.f, 0.f, 0.f, 0.f, 0.f, 0.f, 0.f, 0.f};
  v8f accQ[4] = {zero, zero, zero, zero};
  v8f accK[4] = {zero, zero, zero, zero};
  v8f accV[4] = {zero, zero, zero, zero};

  for (int k0 = 0; k0 < DE; k0 += KC) {
    __syncthreads();
    // Cooperative staging: 64x32 floats per array; coalesced 128B row segments.
#pragma unroll
    for (int i = 0; i < 16; ++i) {
      const int idx = tid + i * 128;          // 0..2047
      const int r = idx >> 5, c = idx & 31;
      xs [r * SX + c] = x [(size_t)(rbase + r) * DE + k0 + c];
      wqs[r * SX + c] = Wq[(size_t)r * DE + k0 + c];
      wks[r * SX + c] = Wk[(size_t)r * DE + k0 + c];
      wvs[r * SX + c] = Wv[(size_t)r * DE + k0 + c];
    }
    __syncthreads();

#pragma unroll
    for (int kc = 0; kc < KC; kc += 4) {
      const int ka = kc + hi2;
      // A fragment: 16x4 slice of x for this wave's 16 rows.
      v2f a;
      a.x = xs[(w * 16 + n15) * SX + ka];
      a.y = xs[(w * 16 + n15) * SX + ka + 1];
#pragma unroll
      for (int t = 0; t < 4; ++t) {
        const int nn = t * 16 + n15;          // output column (B fragment N)
        v2f bq; bq.x = wqs[nn * SX + ka]; bq.y = wqs[nn * SX + ka + 1];
        accQ[t] = wmma_f32(a, bq, accQ[t]);
        v2f bk; bk.x = wks[nn * SX + ka]; bk.y = wks[nn * SX + ka + 1];
        accK[t] = wmma_f32(a, bk, accK[t]);
        v2f bv; bv.x = wvs[nn * SX + ka]; bv.y = wvs[nn * SX + ka + 1];
        accV[t] = wmma_f32(a, bv, accV[t]);
      }
    }
  }

  // Store C/D fragments; fold 1/sqrt(d_k)=0.125 into Q.
#pragma unroll
  for (int t = 0; t < 4; ++t) {
#pragma unroll
    for (int v = 0; v < 8; ++v) {
      const size_t row = (size_t)rbase + w * 16 + v + hi8;
      const int col = t * 16 + n15;
      Qo[row * DH + col] = accQ[t][v] * 0.125f;
      Ko[row * DH + col] = accK[t][v];
      Vo[row * DH + col] = accV[t][v];
    }
  }
}

// ---------------------------------------------------------------------------
// Kernel 2: causal flash attention with the reference's "==0 -> -inf" quirk.
// Block = 128 threads (4 waves) covering 64 query rows of one batch; wave w
// owns 16 rows. K/V 16-row blocks staged in LDS; P round-trips through a
// per-wave padded LDS tile to convert accumulator layout -> A-fragment layout.
// ---------------------------------------------------------------------------
#define PSX 17  // P tile row stride (conflict-free for C/D store + A-frag load)

__global__ __launch_bounds__(128) void flash_attn_kernel(
    const float* __restrict__ Q, const float* __restrict__ K,
    const float* __restrict__ V, float* __restrict__ out)
{
  __shared__ float Ks[16 * SX];
  __shared__ float Vs[16 * SX];
  __shared__ float Ps[4 * 16 * PSX];

  const int tid  = threadIdx.x;
  const int lane = tid & 31;
  const int w    = tid >> 5;
  const int b    = blockIdx.x >> 6;       // SEQ/64 = 64 q-blocks per batch
  const int qb   = blockIdx.x & 63;
  const int n15  = lane & 15;
  const int hi2  = (lane >> 4) << 1;
  const int hi8  = (lane >> 4) << 3;

  const float NEG_INF = -__builtin_inff();
  const size_t qrow0 = (size_t)b * SEQ + (size_t)qb * 64 + w * 16;
  const size_t krow0 = (size_t)b * SEQ;

  // Q held as 16 A-fragments (16x64 tile, already scaled by 0.125).
  v2f qa[16];
#pragma unroll
  for (int kc2 = 0; kc2 < 16; ++kc2) {
    const float* qp = Q + (qrow0 + n15) * DH + kc2 * 4 + hi2;
    qa[kc2].x = qp[0];
    qa[kc2].y = qp[1];
  }

  const v8f zero = {0.f, 0.f, 0.f, 0.f, 0.f, 0.f, 0.f, 0.f};
  v8f o[4] = {zero, zero, zero, zero};
  float mst[8], lst[8];
#pragma unroll
  for (int v = 0; v < 8; ++v) { mst[v] = NEG_INF; lst[v] = 0.f; }

  float* Pw = Ps + w * 16 * PSX;
  const int nkb = qb * 4 + 4;             // causal: keys up to qb*64+63

  for (int kb = 0; kb < nkb; ++kb) {
    __syncthreads();
#pragma unroll
    for (int i = 0; i < 8; ++i) {         // 16x64 floats each, coalesced
      const int idx = tid + i * 128;
      const int r = idx >> 6, c = idx & 63;
      Ks[r * SX + c] = K[(krow0 + kb * 16 + r) * DH + c];
      Vs[r * SX + c] = V[(krow0 + kb * 16 + r) * DH + c];
    }
    __syncthreads();

    // S = Q_tile @ K_block^T  (16x16, K-dim 64 -> 16 WMMAs)
    v8f s = zero;
#pragma unroll
    for (int kc2 = 0; kc2 < 16; ++kc2) {
      const int k = kc2 * 4 + hi2;
      v2f bk; bk.x = Ks[n15 * SX + k]; bk.y = Ks[n15 * SX + k + 1];
      s = wmma_f32(qa[kc2], bk, s);
    }

    // Mask (causal + faithful ==0 quirk), online softmax update.
    const int skey = kb * 16 + n15;
    const int qr0  = qb * 64 + w * 16 + hi8;
    float alpha[8], pv[8];
#pragma unroll
    for (int v = 0; v < 8; ++v) {
      float val = s[v];
      if (skey > qr0 + v || val == 0.0f) val = NEG_INF;
      float r = val;                        // row max across half-wave
      r = fmaxf(r, __shfl_xor(r, 1, 32));
      r = fmaxf(r, __shfl_xor(r, 2, 32));
      r = fmaxf(r, __shfl_xor(r, 4, 32));
      r = fmaxf(r, __shfl_xor(r, 8, 32));
      const float mo = mst[v];
      const float mn = fmaxf(mo, r);
      const float al = (mo == NEG_INF) ? 0.f : __expf(mo - mn);
      const float p  = (val == NEG_INF) ? 0.f : __expf(val - mn);
      float rs = p;                         // row sum
      rs += __shfl_xor(rs, 1, 32);
      rs += __shfl_xor(rs, 2, 32);
      rs += __shfl_xor(rs, 4, 32);
      rs += __shfl_xor(rs, 8, 32);
      lst[v]  = al * lst[v] + rs;
      mst[v]  = mn;
      alpha[v] = al;
      pv[v]    = p;
    }
#pragma unroll
    for (int t = 0; t < 4; ++t)
#pragma unroll
      for (int v = 0; v < 8; ++v)
        o[t][v] *= alpha[v];

    // C/D layout -> row-major LDS tile (per wave).
#pragma unroll
    for (int v = 0; v < 8; ++v)
      Pw[(v + hi8) * PSX + n15] = pv[v];
    __syncthreads();

    // O += P(16x16) @ V_block(16x64): 4 K-steps x 4 N-tiles.
#pragma unroll
    for (int kc2 = 0; kc2 < 4; ++kc2) {
      const int kk = kc2 * 4 + hi2;
      v2f pa; pa.x = Pw[n15 * PSX + kk]; pa.y = Pw[n15 * PSX + kk + 1];
#pragma unroll
      for (int t = 0; t < 4; ++t) {
        v2f bv;
        bv.x = Vs[kk * SX + t * 16 + n15];
        bv.y = Vs[(kk + 1) * SX + t * 16 + n15];
        o[t] = wmma_f32(pa, bv, o[t]);
      }
    }
  }

  // Final normalize + store.
#pragma unroll
  for (int t = 0; t < 4; ++t) {
#pragma unroll
    for (int v = 0; v < 8; ++v) {
      const float inv = 1.0f / lst[v];
      out[(qrow0 + v + hi8) * DH + t * 16 + n15] = o[t][v] * inv;
    }
  }
}

// ---------------------------------------------------------------------------
extern "C" void kernel_launch(void* const* d_in, const int* in_sizes, int n_in,
                              void* d_out, int out_size, void* d_ws, size_t ws_size,
                              hipStream_t stream) {
  (void)in_sizes; (void)n_in; (void)out_size; (void)ws_size;
  const float* x  = (const float*)d_in[0];
  const float* Wq = (const float*)d_in[1];
  const float* Wk = (const float*)d_in[2];
  const float* Wv = (const float*)d_in[3];

  float* Qb = (float*)d_ws;                       // [NROW x 64] scaled Q
  float* Kb = Qb + (size_t)NROW * DH;             // [NROW x 64]
  float* Vb = Kb + (size_t)NROW * DH;             // [NROW x 64]
  float* y  = (float*)d_out;

  qkv_proj_kernel<<<NROW / 64, 128, 0, stream>>>(x, Wq, Wk, Wv, Qb, Kb, Vb);
  flash_attn_kernel<<<BATCH * (SEQ / 64), 128, 0, stream>>>(Qb, Kb, Vb, y);
}